// LinearModel_37160057045302
// MI455X (gfx1250) — compile-verified
//
#include <hip/hip_runtime.h>
#include <hip/hip_bf16.h>

typedef float v2f __attribute__((ext_vector_type(2)));
typedef float v8f __attribute__((ext_vector_type(8)));

#define D_DIM 8192
#define BK 64
#define LDS_STRIDE (BK + 4)          // 68 floats: rows 16B-aligned, conflict-free ds_load_b64
#define ROWS_PER_BLOCK 128
#define WAVES 8
#define THREADS 256
#define KSPLIT 4
#define KCHUNK (D_DIM / KSPLIT)      // 2048
#define NCHUNK (KCHUNK / BK)         // 32
#define NBLOCKS ((D_DIM / ROWS_PER_BLOCK) * KSPLIT)  // 256

// Fused pass: streams W0 (to LDS, then WMMA matvec) and W (registers only, s-dot).
__global__ __launch_bounds__(THREADS) void ntk_main_kernel(
    const float* __restrict__ x, const float* __restrict__ W,
    const float* __restrict__ W0, float* __restrict__ f0p,
    float* __restrict__ spart)
{
  __shared__ float lds_w0[ROWS_PER_BLOCK * LDS_STRIDE];
  __shared__ float lds_x[BK];
  __shared__ float lds_red[WAVES];

  const int tid    = threadIdx.x;
  const int lane   = tid & 31;
  const int wave   = tid >> 5;
  const int bx     = blockIdx.x;
  const int kslice = bx & (KSPLIT - 1);
  const int rowBase = (bx >> 2) * ROWS_PER_BLOCK;
  const int k0base  = kslice * KCHUNK;

  v8f   acc   = {};     // D[m][n]: f0 partial for row m, replicated over n
  float s_acc = 0.0f;   // per-thread partial of dot(W - W0, W0)

  const int rA   = (wave << 4) + (lane & 15);  // A-matrix row (M) for this lane
  const int cSel = (lane >> 4) << 1;           // K sub-offset: 0 (lanes 0-15) / 2 (lanes 16-31)

  for (int c = 0; c < NCHUNK; ++c) {
    const int k0 = k0base + c * BK;

    // ---- stage W0 tile (128 x 64) to LDS; consume W on the fly for the s-dot ----
    #pragma unroll
    for (int i = 0; i < 8; ++i) {
      int linear = i * THREADS + tid;       // 2048 float4 slots
      int row = linear >> 4;                // 16 float4 per row
      int col = (linear & 15) << 2;
      size_t g = (size_t)(rowBase + row) * D_DIM + (size_t)(k0 + col);
      float4 w0v = *(const float4*)(W0 + g);
      float4 wv  = *(const float4*)(W  + g);
      s_acc += (wv.x - w0v.x) * w0v.x;
      s_acc += (wv.y - w0v.y) * w0v.y;
      s_acc += (wv.z - w0v.z) * w0v.z;
      s_acc += (wv.w - w0v.w) * w0v.w;
      *(float4*)(&lds_w0[row * LDS_STRIDE + col]) = w0v;
    }
    if (tid < BK) lds_x[tid] = x[k0 + tid];
    __syncthreads();

    // prefetch next tile into cache hierarchy (global_prefetch_b8)
    if (c + 1 < NCHUNK) {
      size_t gp = (size_t)(rowBase + (tid >> 1)) * D_DIM + (size_t)(k0 + BK);
      __builtin_prefetch(W0 + gp, 0, 0);
      __builtin_prefetch(W  + gp, 0, 0);
    }

    // ---- WMMA matvec: 16 rows per wave, K advances 4 per v_wmma_f32_16x16x4_f32 ----
    #pragma unroll
    for (int kk = 0; kk < BK; kk += 4) {
      // A 16x4 layout (§7.12.2): lane m, VGPR0/1 = K {cSel, cSel+1}
      v2f a = *(const v2f*)(&lds_w0[rA * LDS_STRIDE + kk + cSel]);
      // B 4x16: column n = x[k] broadcast (all lanes in half read same LDS word)
      v2f b;
      b.x = lds_x[kk + cSel];
      b.y = lds_x[kk + cSel + 1];
      acc = __builtin_amdgcn_wmma_f32_16x16x4_f32(
          /*neg_a=*/false, a, /*neg_b=*/false, b,
          /*c_mod=*/(short)0, acc, /*reuse_a=*/false, /*reuse_b=*/false);
    }
    __syncthreads();
  }

  // ---- write f0 partials: lanes 0 and 16 hold column N=0 of D ----
  if (lane == 0 || lane == 16) {
    int mbase = (lane >> 4) << 3;   // lane 0 -> M=0..7, lane 16 -> M=8..15
    #pragma unroll
    for (int j = 0; j < 8; ++j) {
      int row = rowBase + (wave << 4) + mbase + j;
      f0p[(size_t)kslice * D_DIM + row] = acc[j];
    }
  }

  // ---- deterministic block reduction of s partials ----
  for (int off = 16; off > 0; off >>= 1)
    s_acc += __shfl_down(s_acc, off, 32);
  if (lane == 0) lds_red[wave] = s_acc;
  __syncthreads();
  if (wave == 0) {
    float v = (lane < WAVES) ? lds_red[lane] : 0.0f;
    for (int off = 4; off > 0; off >>= 1)
      v += __shfl_down(v, off, 32);
    if (lane == 0) spart[bx] = v;
  }
}

// Final: s = sum of 256 block partials (fixed order), f[r] = sum_k f0p[k][r] + s.
__global__ __launch_bounds__(THREADS) void ntk_finish_kernel(
    const float* __restrict__ f0p, const float* __restrict__ spart,
    float* __restrict__ out)
{
  __shared__ float red[WAVES];
  __shared__ float s_total;
  const int tid = threadIdx.x, lane = tid & 31, wave = tid >> 5;

  float v = spart[tid];                      // exactly NBLOCKS==THREADS partials
  for (int off = 16; off > 0; off >>= 1)
    v += __shfl_down(v, off, 32);
  if (lane == 0) red[wave] = v;
  __syncthreads();
  if (tid == 0) {
    float s = 0.0f;
    for (int i = 0; i < WAVES; ++i) s += red[i];
    s_total = s;
  }
  __syncthreads();
  const float s = s_total;

  for (int r = tid; r < D_DIM; r += THREADS) {
    float f = f0p[r] + f0p[D_DIM + r] + f0p[2 * D_DIM + r] + f0p[3 * D_DIM + r];
    out[r] = f + s;
  }
}

extern "C" void kernel_launch(void* const* d_in, const int* in_sizes, int n_in,
                              void* d_out, int out_size, void* d_ws, size_t ws_size,
                              hipStream_t stream) {
  const float* x  = (const float*)d_in[0];
  const float* W  = (const float*)d_in[1];
  const float* W0 = (const float*)d_in[2];

  float* f0p   = (float*)d_ws;                      // KSPLIT * D_DIM floats
  float* spart = f0p + (size_t)KSPLIT * D_DIM;      // NBLOCKS floats

  ntk_main_kernel<<<NBLOCKS, THREADS, 0, stream>>>(x, W, W0, f0p, spart);
  ntk_finish_kernel<<<1, THREADS, 0, stream>>>(f0p, spart, (float*)d_out);
}